// BottomUpChildSumTreeLSTM_54039278518891
// MI455X (gfx1250) — compile-verified
//
#include <hip/hip_runtime.h>

// ---------------------------------------------------------------------------
// Bottom-up child-sum TreeLSTM for MI455X (gfx1250), fp32 WMMA path.
// B = N = D = U = 256.
//
// ws layout (floats):
//   [0,            67108864)  fiou_xb  : (B*N, 1024) = inputs @ x_fiou + bias
//   [67108864,     83886080)  csh      : (B, N, U) child-sum hidden accum
//   [83886080,    100663296)  gcs      : (B, N, U) gated cell accum
//
// Tensor Data Mover stages Wf (256x256 f32, 256 KB) into LDS once per
// persistent scan block, with TDM pad_enable giving a 260-float row stride
// (bank-conflict-free B reads). Removes ~1 GB of per-step L2 traffic from
// the serial critical path.
// ---------------------------------------------------------------------------

typedef __attribute__((ext_vector_type(2))) float v2f;
typedef __attribute__((ext_vector_type(8))) float v8f;
typedef __attribute__((ext_vector_type(4))) unsigned int v4u;
typedef __attribute__((ext_vector_type(8))) int v8i;
typedef __attribute__((ext_vector_type(4))) int v4i;

#define BQ 256
#define NN 256
#define DD 256
#define UU 256

#define WF_STRIDE 260              // 256 + TDM pad (4 DWORDs per 256 DWORDs)
#define WF_LDS_BYTES (256 * WF_STRIDE * 4)   // 266,240 B dynamic LDS

__device__ __forceinline__ float sigmoidf_(float x) {
    return 1.0f / (1.0f + __expf(-x));
}

// V_WMMA_F32_16X16X4_F32:  D(16x16,f32) = A(16x4,f32) * B(4x16,f32) + C
// A frag: lane l holds row M=l&15, k-pair base 2*(l>>4) -> {a.x=K+0, a.y=K+1}
// B frag: lane l holds col N=l&15, k rows {2*(l>>4), 2*(l>>4)+1}
// C/D   : vgpr i, lane l -> M = i + 8*(l>>4), N = l&15
__device__ __forceinline__ v8f wmma_k4(v2f a, v2f b, v8f c) {
    return __builtin_amdgcn_wmma_f32_16x16x4_f32(
        /*neg_a=*/false, a, /*neg_b=*/false, b,
        /*c_mod=*/(short)0, c, /*reuse_a=*/false, /*reuse_b=*/false);
}

// ---------------------------------------------------------------------------
__global__ void zero_kernel(float4* __restrict__ p, long long n4) {
    long long i = (long long)blockIdx.x * blockDim.x + threadIdx.x;
    long long s = (long long)gridDim.x * blockDim.x;
    float4 z = {0.f, 0.f, 0.f, 0.f};
    for (; i < n4; i += s) p[i] = z;
}

// ---------------------------------------------------------------------------
// fiou_xb[m, c] = sum_k inputs[m,k] * Wx[k,c] + bias[c]
// grid.x = (B*N)/16 = 4096 blocks of 256 threads (8 waves).
// ---------------------------------------------------------------------------
__global__ __launch_bounds__(256) void xproj_kernel(
    const float* __restrict__ X,     // (65536, 256)
    const float* __restrict__ Wx,    // (256, 1024)
    const float* __restrict__ bias,  // (1024)
    float* __restrict__ out)         // (65536, 1024)
{
    __shared__ float sA[16][260];    // padded: stride % 64 == 4 -> conflict-free

    const int m0  = blockIdx.x * 16;
    const int tid = threadIdx.x;

    {   // cooperative A-tile load: 16 floats per thread (4x float4)
        const int r = tid >> 4;
        const int c = (tid & 15) * 16;
        const float4* src = (const float4*)(X + (size_t)(m0 + r) * DD + c);
        float4* dst = (float4*)(&sA[r][c]);
        dst[0] = src[0]; dst[1] = src[1]; dst[2] = src[2]; dst[3] = src[3];
    }
    __syncthreads();

    const int lane = tid & 31;
    const int wave = tid >> 5;
    const int half = lane >> 4;
    const int n    = lane & 15;      // M for A-frag, N for B/C-frag
    const int kb   = half * 2;

    for (int j = 0; j < 8; ++j) {
        const int col0 = (wave + j * 8) * 16;
        v8f acc = {0.f, 0.f, 0.f, 0.f, 0.f, 0.f, 0.f, 0.f};
        for (int kk = 0; kk < DD; kk += 4) {
            v2f a, b;
            a.x = sA[n][kk + kb];
            a.y = sA[n][kk + kb + 1];
            b.x = Wx[(size_t)(kk + kb)     * 1024 + col0 + n];
            b.y = Wx[(size_t)(kk + kb + 1) * 1024 + col0 + n];
            acc = wmma_k4(a, b, acc);
        }
#pragma unroll
        for (int i = 0; i < 8; ++i) {
            const int row = m0 + i + half * 8;
            const int col = col0 + n;
            out[(size_t)row * 1024 + col] = acc[i] + bias[col];
        }
    }
}

// ---------------------------------------------------------------------------
// Persistent scan: 16 blocks, each owns 16 batch rows, runs 256 steps.
// Wf is TDM-staged into (dynamic) LDS once, padded to stride 260.
// ---------------------------------------------------------------------------
__global__ __launch_bounds__(256) void tree_scan_kernel(
    const int*   __restrict__ parents,      // (B, N)
    const int*   __restrict__ post_orders,  // (B, N)
    const float* __restrict__ Wiou,         // (256, 768)
    const float* __restrict__ Wf,           // (256, 256)
    const float* __restrict__ fiou,         // (B*N, 1024)
    float* __restrict__ csh,                // (B, N, U)
    float* __restrict__ gcs,                // (B, N, U)
    float* __restrict__ hs)                 // (B, N, U) = d_out
{
    extern __shared__ float sWf[];          // 256 x 260 padded copy of Wf

    __shared__ float sA[16][260];           // csh[b, tgt, :] gather (GEMM1 A)
    __shared__ float sOut[16][260];         // out_t             (GEMM2 A)
    __shared__ float sMem[16][260];         // mem_t
    __shared__ int   sTgt[16];
    __shared__ int   sPar[16];
    __shared__ float sOmf[16];
    __shared__ float sPmf[16];

    const int r0   = blockIdx.x * 16;       // first batch row of this WG
    const int tid  = threadIdx.x;
    const int lane = tid & 31;
    const int wave = tid >> 5;
    const int half = lane >> 4;
    const int n    = lane & 15;
    const int kb   = half * 2;

    // ---- TDM: stage Wf (256x256 f32) Global -> LDS, padded rows ----
    if (wave == 0) {
        unsigned int lds_base = (unsigned int)(unsigned long long)(&sWf[0]);
        unsigned long long ga = (unsigned long long)Wf;

        v4u g0;
        g0[0] = 1u;                                   // count=1 (valid, user)
        g0[1] = lds_base;                             // lds_addr (bytes)
        g0[2] = (unsigned int)(ga & 0xffffffffull);   // global_addr[31:0]
        g0[3] = (unsigned int)((ga >> 32) & 0x1ffffffull)
              | (2u << 30);                           // addr[56:32] | type=2

        v8i g1;
        g1[0] = (int)((2u << 16)     // data_size = 2 -> 4-byte elements
                    | (1u << 20)     // pad_enable
                    | (7u << 22)     // pad_interval = 7 -> every 256 DWORDs
                    | (3u << 25));   // pad_amount   = 3 -> 4 DWORDs pad
        g1[1] = (int)(256u << 16);   // tensor_dim0[15:0] = 256
        g1[2] = (int)(256u << 16);   // tensor_dim1[15:0] = 256
        g1[3] = (int)(256u << 16);   // tile_dim0 = 256
        g1[4] = (int)(256u);         // tile_dim1 = 256, tile_dim2 = 0
        g1[5] = (int)(256u);         // tensor_dim0_stride[31:0] = 256
        g1[6] = 0;
        g1[7] = 0;

        v4i z4 = {0, 0, 0, 0};                 // 2-D tensor: groups 2/3 unused
        v8i z8 = {0, 0, 0, 0, 0, 0, 0, 0};     // extra group (6-arg form)
        __builtin_amdgcn_tensor_load_to_lds(g0, g1, z4, z4, z8, 0);
        __builtin_amdgcn_s_wait_tensorcnt(0);
    }
    __syncthreads();

    for (int t = 0; t < NN; ++t) {
        // ---- step metadata (one thread per batch row) ----
        if (tid < 16) {
            const int b    = r0 + tid;
            const int pov  = post_orders[b * NN + t];
            const int om   = pov >= 0;
            const int tgt  = om ? pov : 0;
            const int praw = parents[b * NN + tgt];
            const int pm   = (praw >= 0) && om;
            sTgt[tid] = tgt;
            sPar[tid] = praw >= 0 ? praw : 0;
            sOmf[tid] = om ? 1.0f : 0.0f;
            sPmf[tid] = pm ? 1.0f : 0.0f;
        }
        __syncthreads();

        // ---- gather csh[b, tgt, :] into LDS ----
        {
            const int r = tid >> 4;
            const int c = (tid & 15) * 16;
            const float4* src = (const float4*)(
                csh + ((size_t)(r0 + r) * NN + sTgt[r]) * UU + c);
            float4* dst = (float4*)(&sA[r][c]);
            dst[0] = src[0]; dst[1] = src[1]; dst[2] = src[2]; dst[3] = src[3];
        }
        __syncthreads();

        // ---- GEMM1: iou_h = csh_tile @ Wiou  (16x256)@(256x768) ----
        // wave w owns column tiles {w, w+8, w+16, w+24, w+32, w+40}
        // -> q=0,1: i-gate cols, q=2,3: o-gate, q=4,5: u-gate, column-aligned.
        v8f acc[6];
#pragma unroll
        for (int q = 0; q < 6; ++q)
            acc[q] = (v8f){0.f, 0.f, 0.f, 0.f, 0.f, 0.f, 0.f, 0.f};

        for (int kk = 0; kk < UU; kk += 4) {
            v2f a;
            a.x = sA[n][kk + kb];
            a.y = sA[n][kk + kb + 1];
#pragma unroll
            for (int q = 0; q < 6; ++q) {
                const int col = (wave + q * 8) * 16 + n;
                v2f b;
                b.x = Wiou[(size_t)(kk + kb)     * 768 + col];
                b.y = Wiou[(size_t)(kk + kb + 1) * 768 + col];
                acc[q] = wmma_k4(a, b, acc[q]);
            }
        }

        // ---- gates / cell / hidden ----
#pragma unroll
        for (int g = 0; g < 2; ++g) {
            const int colr = (wave + g * 8) * 16 + n;   // in [0, 256)
            v8f iv = acc[g], ov = acc[2 + g], uv = acc[4 + g];
#pragma unroll
            for (int i = 0; i < 8; ++i) {
                const int M = i + half * 8;
                const int b = r0 + M;
                const size_t node = (size_t)b * NN + sTgt[M];
                const float xi = iv[i] + fiou[node * 1024 + 256 + colr];
                const float xo = ov[i] + fiou[node * 1024 + 512 + colr];
                const float xu = uv[i] + fiou[node * 1024 + 768 + colr];
                const float mem  = sigmoidf_(xi) * tanhf(xu) + gcs[node * UU + colr];
                const float outv = sigmoidf_(xo) * tanhf(mem);
                sMem[M][colr] = mem;
                sOut[M][colr] = outv;
                hs[node * UU + colr] += outv * sOmf[M];
            }
        }
        __syncthreads();

        // ---- GEMM2: fg = out_t @ Wf  (16x256)@(256x256), Wf from LDS ----
        v8f acc2[2];
        acc2[0] = (v8f){0.f, 0.f, 0.f, 0.f, 0.f, 0.f, 0.f, 0.f};
        acc2[1] = (v8f){0.f, 0.f, 0.f, 0.f, 0.f, 0.f, 0.f, 0.f};
        for (int kk = 0; kk < UU; kk += 4) {
            v2f a;
            a.x = sOut[n][kk + kb];
            a.y = sOut[n][kk + kb + 1];
#pragma unroll
            for (int q = 0; q < 2; ++q) {
                const int col = (wave + q * 8) * 16 + n;
                v2f b;
                b.x = sWf[(kk + kb)     * WF_STRIDE + col];
                b.y = sWf[(kk + kb + 1) * WF_STRIDE + col];
                acc2[q] = wmma_k4(a, b, acc2[q]);
            }
        }

        // ---- f-gate, scatter-add into parent's csh / gcs ----
#pragma unroll
        for (int q = 0; q < 2; ++q) {
            const int colr = (wave + q * 8) * 16 + n;
#pragma unroll
            for (int i = 0; i < 8; ++i) {
                const int M = i + half * 8;
                const int b = r0 + M;
                const size_t par = (size_t)b * NN + sPar[M];
                const float x     = acc2[q][i] + fiou[par * 1024 + colr]; // f_xb
                const float gated = sigmoidf_(x) * sMem[M][colr];
                const float pmf   = sPmf[M];
                csh[par * UU + colr] += sOut[M][colr] * pmf;
                gcs[par * UU + colr] += gated * pmf;
            }
        }

        __threadfence();   // make csh/gcs/hs RMWs visible before next step
        __syncthreads();
    }
}

// ---------------------------------------------------------------------------
extern "C" void kernel_launch(void* const* d_in, const int* in_sizes, int n_in,
                              void* d_out, int out_size, void* d_ws, size_t ws_size,
                              hipStream_t stream) {
    const float* inputs      = (const float*)d_in[0];  // (B, N, D)
    const int*   parents     = (const int*)  d_in[1];  // (B, N)
    const int*   post_orders = (const int*)  d_in[2];  // (B, N)
    const float* Wx          = (const float*)d_in[3];  // (D, 4U)
    const float* Wf          = (const float*)d_in[4];  // (U, U)
    const float* Wiou        = (const float*)d_in[5];  // (U, 3U)
    const float* bias        = (const float*)d_in[6];  // (4U)

    float* hs   = (float*)d_out;
    float* ws   = (float*)d_ws;
    float* fiou = ws;                      // 67,108,864 floats
    float* csh  = ws + 67108864ull;        // 16,777,216 floats
    float* gcs  = ws + 83886080ull;        // 16,777,216 floats

    // zero csh+gcs (contiguous) and hs
    zero_kernel<<<8192, 256, 0, stream>>>((float4*)csh, 33554432ll / 4);
    zero_kernel<<<4096, 256, 0, stream>>>((float4*)hs,  16777216ll / 4);

    // phase 1: x-projection GEMM
    xproj_kernel<<<(BQ * NN) / 16, 256, 0, stream>>>(inputs, Wx, bias, fiou);

    // phase 2: persistent per-batch-group scan (dynamic LDS holds padded Wf)
    tree_scan_kernel<<<BQ / 16, 256, WF_LDS_BYTES, stream>>>(
        parents, post_orders, Wiou, Wf, fiou, csh, gcs, hs);
}